// AssistedExcitation_66494683676730
// MI455X (gfx1250) — compile-verified
//
#include <hip/hip_runtime.h>

typedef __attribute__((ext_vector_type(2))) float v2f;
typedef __attribute__((ext_vector_type(4))) float v4f;
typedef __attribute__((ext_vector_type(8))) float v8f;

#define B_  16
#define C_  256
#define H_  80
#define W_  80
#define NBOX_ 32
#define HW_ (H_ * W_)          // 6400
#define TOT_ (B_ * C_ * HW_)   // 26214400

// ---------------------------------------------------------------------------
// Kernel 1: build the 80x80 reflect-padded 1D-convolution band matrix G.
// G[o][i] = sum_{d=-2..2} w[d] * [refl(o+d) == i], w derived from the 5x5
// kernel input (k2d = outer(w,w), so w[j] = k2d[2][j] / sqrt(k2d[2][2])).
// ---------------------------------------------------------------------------
__global__ void ae_build_g(const float* __restrict__ k2d, float* __restrict__ G) {
  int i = blockIdx.x * blockDim.x + threadIdx.x;
  if (i >= H_ * H_) return;
  int o = i / H_, col = i % H_;
  float w2 = sqrtf(k2d[12]);                  // center element k2d[2][2]
  float s = 0.f;
#pragma unroll
  for (int d = -2; d <= 2; ++d) {
    float wd = k2d[10 + (d + 2)] / w2;        // row 2 of 5x5 kernel
    int r = o + d;
    if (r < 0) r = -r;                        // reflect (no edge repeat)
    if (r > H_ - 1) r = 2 * (H_ - 1) - r;
    if (r == col) s += wd;
  }
  G[i] = s;
}

// ---------------------------------------------------------------------------
// Kernel 2: rasterize union of boxes into binary mask M[b][h][w].
// ---------------------------------------------------------------------------
__global__ void ae_rasterize(const float* __restrict__ boxes, float* __restrict__ M) {
  int idx = blockIdx.x * blockDim.x + threadIdx.x;
  if (idx >= B_ * HW_) return;
  int b = idx / HW_;
  int hw = idx % HW_;
  int h = hw / W_, w = hw % W_;
  const float* bx = boxes + b * NBOX_ * 4;
  float m = 0.f;
#pragma unroll 4
  for (int n = 0; n < NBOX_; ++n) {
    float xc = bx[n * 4 + 0], yc = bx[n * 4 + 1];
    float bw = bx[n * 4 + 2], bh = bx[n * 4 + 3];
    int x1 = (int)fmaxf(0.f, truncf((xc - bw * 0.5f) * (float)W_));
    int y1 = (int)fmaxf(0.f, truncf((yc - bh * 0.5f) * (float)H_));
    int x2 = (int)fminf((float)(W_ - 1), truncf((xc + bw * 0.5f) * (float)W_));
    int y2 = (int)fminf((float)(H_ - 1), truncf((yc + bh * 0.5f) * (float)H_));
    bool hit = (x2 > x1) && (y2 > y1) &&
               (h >= y1) && (h <= y2) && (w >= x1) && (w <= x2);
    if (hit) m = 1.f;
  }
  M[idx] = m;
}

// ---------------------------------------------------------------------------
// WMMA helper: one wave computes a 16x16 f32 tile of D = A(80xk) * B(kx80)
// using V_WMMA_F32_16X16X4_F32.  Layout (CDNA5 ISA 7.12.2):
//   A 16x4 : lane l -> row l&15, k = 2*(l>>4) + vgpr
//   B 4x16 : lane l -> col l&15, k = 2*(l>>4) + vgpr
//   D      : VGPR j -> row j + 8*(l>>4), col l&15
// ---------------------------------------------------------------------------

// Stage 1: T[b] = M[b] (80x80) * G^T (80x80);  Bop[k][n] = G[(tx+n)*80 + (c0+k)]
__global__ void ae_stage1(const float* __restrict__ M, const float* __restrict__ G,
                          float* __restrict__ T) {
  int wave = blockIdx.x;              // 400 waves: 16 images x 25 tiles
  int b = wave / 25;
  int tile = wave % 25;
  int ty = (tile / 5) * 16;
  int tx = (tile % 5) * 16;
  int lane = threadIdx.x;
  int half = lane >> 4;
  int lm = lane & 15;
  const float* Mb = M + b * HW_;
  v8f acc = {};
  for (int c0 = 0; c0 < W_; c0 += 4) {
    int k = c0 + 2 * half;
    v2f a, bo;
    a.x = Mb[(ty + lm) * W_ + k];
    a.y = Mb[(ty + lm) * W_ + k + 1];
    bo.x = G[(tx + lm) * W_ + k];       // G^T tile
    bo.y = G[(tx + lm) * W_ + k + 1];
    acc = __builtin_amdgcn_wmma_f32_16x16x4_f32(false, a, false, bo,
                                                (short)0, acc, false, false);
  }
  float* Tb = T + b * HW_;
#pragma unroll
  for (int j = 0; j < 8; ++j)
    Tb[(ty + j + 8 * half) * W_ + tx + lm] = acc[j];
}

// Stage 2: scale[b] = 1 + G (80x80) * T[b] (80x80)
__global__ void ae_stage2(const float* __restrict__ G, const float* __restrict__ T,
                          float* __restrict__ scale) {
  int wave = blockIdx.x;
  int b = wave / 25;
  int tile = wave % 25;
  int ty = (tile / 5) * 16;
  int tx = (tile % 5) * 16;
  int lane = threadIdx.x;
  int half = lane >> 4;
  int lm = lane & 15;
  const float* Tb = T + b * HW_;
  v8f acc = {};
  for (int r0 = 0; r0 < H_; r0 += 4) {
    int k = r0 + 2 * half;
    v2f a, bo;
    a.x = G[(ty + lm) * H_ + k];
    a.y = G[(ty + lm) * H_ + k + 1];
    bo.x = Tb[k * W_ + tx + lm];
    bo.y = Tb[(k + 1) * W_ + tx + lm];
    acc = __builtin_amdgcn_wmma_f32_16x16x4_f32(false, a, false, bo,
                                                (short)0, acc, false, false);
  }
  float* Sb = scale + b * HW_;
#pragma unroll
  for (int j = 0; j < 8; ++j)
    Sb[(ty + j + 8 * half) * W_ + tx + lm] = 1.0f + acc[j];
}

// ---------------------------------------------------------------------------
// Kernel 5: bandwidth stage. out = x * scale (scale broadcast over channels).
// float4 streaming with non-temporal hints on the once-touched 105MB arrays.
// ---------------------------------------------------------------------------
__global__ void ae_apply(const float* __restrict__ x, const float* __restrict__ scale,
                         float* __restrict__ out) {
  long long t = (long long)blockIdx.x * blockDim.x + threadIdx.x;
  if (t >= (long long)(TOT_ / 4)) return;
  long long e = t * 4;                      // W=80 divisible by 4: no row crossing
  int b = (int)(e / ((long long)C_ * HW_));
  int hw = (int)(e % HW_);
  v4f xv = __builtin_nontemporal_load((const v4f*)(x + e));
  v4f sv = *(const v4f*)(scale + (long long)b * HW_ + hw);   // tiny, cache it
  v4f ov = xv * sv;
  __builtin_nontemporal_store(ov, (v4f*)(out + e));
}

// ---------------------------------------------------------------------------
extern "C" void kernel_launch(void* const* d_in, const int* in_sizes, int n_in,
                              void* d_out, int out_size, void* d_ws, size_t ws_size,
                              hipStream_t stream) {
  const float* x     = (const float*)d_in[0];   // [16,256,80,80]
  const float* boxes = (const float*)d_in[1];   // [16,32,4]
  const float* k2d   = (const float*)d_in[2];   // [1,1,5,5]
  float* out = (float*)d_out;

  float* ws = (float*)d_ws;
  float* G     = ws;                    //  6400 f32
  float* M     = ws + 6400;             // 102400 f32
  float* T     = ws + 6400 + 102400;    // 102400 f32
  float* scale = ws + 6400 + 204800;    // 102400 f32  (~1.23 MB total)

  ae_build_g<<<(H_ * H_ + 255) / 256, 256, 0, stream>>>(k2d, G);
  ae_rasterize<<<(B_ * HW_ + 255) / 256, 256, 0, stream>>>(boxes, M);
  ae_stage1<<<B_ * 25, 32, 0, stream>>>(M, G, T);
  ae_stage2<<<B_ * 25, 32, 0, stream>>>(G, T, scale);
  ae_apply<<<(TOT_ / 4 + 255) / 256, 256, 0, stream>>>(x, scale, out);
}